// HybridGNN_31662498906635
// MI455X (gfx1250) — compile-verified
//
#include <hip/hip_runtime.h>
#include <hip/hip_bf16.h>

namespace {

constexpr int IN_F  = 128;
constexpr int HID_C = 320;
constexpr int HEADS_C = 4;
constexpr int DH_C  = 80;
constexpr float NEG_SLOPE_C = 0.2f;
constexpr float EPS_C = 1e-5f;

constexpr int LDS_PAD = 4;                    // dword pad -> conflict-free frag reads
constexpr int LDS_W   = HID_C + LDS_PAD;      // max padded row (K<=320)

typedef float v2f __attribute__((ext_vector_type(2)));
typedef float v8f __attribute__((ext_vector_type(8)));

__device__ __forceinline__ float leakyf(float x) { return x >= 0.f ? x : NEG_SLOPE_C * x; }

// order-preserving float<->uint encoding for atomicMax on floats
__device__ __forceinline__ unsigned fenc(float x) {
  unsigned u = __float_as_uint(x);
  return (u & 0x80000000u) ? ~u : (u | 0x80000000u);
}
__device__ __forceinline__ float fdec(unsigned u) {
  u = (u & 0x80000000u) ? (u & 0x7fffffffu) : ~u;
  return __uint_as_float(u);
}

// Cooperative, coalesced load of a 16 x K fp32 tile (row stride K) into padded LDS.
// 16*K/4 float4 chunks, divisible by 32 for K in {128,320} -> no lane divergence.
__device__ __forceinline__ void load_tileA(const float* __restrict__ A, float* __restrict__ As,
                                           int K, int ldas) {
  const int lane = threadIdx.x & 31;
  const int kq = K >> 2;
  const int nch = 16 * kq;
  for (int t = lane; t < nch; t += 32) {
    int row = t / kq, c4 = t - row * kq;
    float4 v = *(const float4*)(A + row * K + c4 * 4);
    *(float4*)(As + row * ldas + c4 * 4) = v;
  }
}

// Accumulate NT adjacent 16x16 tiles over K with V_WMMA_F32_16X16X4_F32.
// A fragment comes from LDS (one ds_load_b64 per K-step feeds NT WMMAs).
// Per ISA 7.12.2: A lane l holds row (l&15), K pair 2*(l>>4); B lane l holds
// col (l&15), same K pair; C/D lane l vgpr v -> M = v + 8*(l>>4), N = l&15.
template <int NT>
__device__ __forceinline__ void wmma_accum(const float* __restrict__ As, int ldas,
                                           const float* __restrict__ B, int ldb,
                                           int K, v8f acc[NT]) {
  const int lane = threadIdx.x & 31;
  const int half = lane >> 4;
  const int idx  = lane & 15;
  const float* ap = As + idx * ldas + 2 * half;
  const float* bp = B + idx + 2 * half * ldb;
  #pragma unroll 4
  for (int k = 0; k < K; k += 4) {
    v2f a = *(const v2f*)(ap + k);
    #pragma unroll
    for (int t = 0; t < NT; ++t) {
      v2f b;
      b.x = bp[k * ldb + t * 16];
      b.y = bp[(k + 1) * ldb + t * 16];
      acc[t] = __builtin_amdgcn_wmma_f32_16x16x4_f32(false, a, false, b, (short)0,
                                                     acc[t], false, false);
    }
  }
}

// ---------------- GEMM kernels (one wave per 16 x (16*NT) strip) ----------------

// C = A @ B  (A: [M,K], B: [K,NC], C: [M,NC])
template <int NT>
__global__ __launch_bounds__(32) void k_gemm_plain(const float* __restrict__ A,
                                                   const float* __restrict__ B,
                                                   float* __restrict__ C,
                                                   int K, int NC) {
  __shared__ float As[16 * LDS_W];
  const int m0 = blockIdx.x * 16, n0 = blockIdx.y * (16 * NT);
  const int ldas = K + LDS_PAD;
  load_tileA(A + m0 * K, As, K, ldas);
  __syncthreads();
  v8f acc[NT] = {};
  wmma_accum<NT>(As, ldas, B + n0, NC, K, acc);
  const int lane = threadIdx.x & 31, half = lane >> 4, col = lane & 15;
  #pragma unroll
  for (int t = 0; t < NT; ++t) {
    float* Cp = C + m0 * NC + n0 + t * 16 + col;
    #pragma unroll
    for (int v = 0; v < 8; ++v) Cp[(v + 8 * half) * NC] = acc[t][v];
  }
}

// C = relu(A @ B + bias)
template <int NT>
__global__ __launch_bounds__(32) void k_gemm_bias_relu(const float* __restrict__ A,
                                                       const float* __restrict__ B,
                                                       const float* __restrict__ bias,
                                                       float* __restrict__ C,
                                                       int K, int NC) {
  __shared__ float As[16 * LDS_W];
  const int m0 = blockIdx.x * 16, n0 = blockIdx.y * (16 * NT);
  const int ldas = K + LDS_PAD;
  load_tileA(A + m0 * K, As, K, ldas);
  __syncthreads();
  v8f acc[NT] = {};
  wmma_accum<NT>(As, ldas, B + n0, NC, K, acc);
  const int lane = threadIdx.x & 31, half = lane >> 4, col = lane & 15;
  #pragma unroll
  for (int t = 0; t < NT; ++t) {
    const float bb = bias[n0 + t * 16 + col];
    float* Cp = C + m0 * NC + n0 + t * 16 + col;
    #pragma unroll
    for (int v = 0; v < 8; ++v) Cp[(v + 8 * half) * NC] = fmaxf(acc[t][v] + bb, 0.f);
  }
}

// SAGE combine: C = relu(bn(A1@B1 + A2@B2 + bl))  (K = NC = HID)
template <int NT>
__global__ __launch_bounds__(32) void k_gemm_sage(const float* __restrict__ A1,
                                                  const float* __restrict__ B1,
                                                  const float* __restrict__ A2,
                                                  const float* __restrict__ B2,
                                                  const float* __restrict__ bl,
                                                  const float* __restrict__ gamma,
                                                  const float* __restrict__ beta,
                                                  float* __restrict__ C) {
  __shared__ float As[16 * LDS_W];
  const int m0 = blockIdx.x * 16, n0 = blockIdx.y * (16 * NT);
  const int ldas = HID_C + LDS_PAD;
  v8f acc[NT] = {};
  load_tileA(A1 + m0 * HID_C, As, HID_C, ldas);
  __syncthreads();
  wmma_accum<NT>(As, ldas, B1 + n0, HID_C, HID_C, acc);
  __syncthreads();
  load_tileA(A2 + m0 * HID_C, As, HID_C, ldas);
  __syncthreads();
  wmma_accum<NT>(As, ldas, B2 + n0, HID_C, HID_C, acc);
  const int lane = threadIdx.x & 31, half = lane >> 4, col = lane & 15;
  #pragma unroll
  for (int t = 0; t < NT; ++t) {
    const int c = n0 + t * 16 + col;
    const float g = gamma[c] * rsqrtf(1.0f + EPS_C);
    const float bt = beta[c];
    const float bb = bl[c];
    float* Cp = C + m0 * HID_C + c;
    #pragma unroll
    for (int v = 0; v < 8; ++v) {
      float x = (acc[t][v] + bb) * g + bt;
      Cp[(v + 8 * half) * HID_C] = fmaxf(x, 0.f);
    }
  }
}

// ---------------- elementwise / edge kernels ----------------

__global__ void k_fill(float* __restrict__ p, float v, int n) {
  int i = blockIdx.x * blockDim.x + threadIdx.x;
  if (i < n) p[i] = v;
}

__global__ void k_deg(const int* __restrict__ dst, float* __restrict__ deg, int E) {
  int e = blockIdx.x * blockDim.x + threadIdx.x;
  if (e < E) atomicAdd(&deg[dst[e]], 1.0f);
}

// scatter-sum: S[dst[e], :] += X[src[e], :]   (thread = (edge, 4-channel chunk))
__global__ void k_scatter(const float* __restrict__ X, const int* __restrict__ src,
                          const int* __restrict__ dst, float* __restrict__ S, int E) {
  int t = blockIdx.x * blockDim.x + threadIdx.x;
  if (t >= E * (HID_C / 4)) return;
  int e = t / (HID_C / 4), c4 = t % (HID_C / 4);
  const float4 v = *(const float4*)(X + src[e] * HID_C + c4 * 4);
  float* o = S + dst[e] * HID_C + c4 * 4;
  atomicAdd(o + 0, v.x);
  atomicAdd(o + 1, v.y);
  atomicAdd(o + 2, v.z);
  atomicAdd(o + 3, v.w);
}

__global__ void k_mean(float* __restrict__ S, const float* __restrict__ deg, int n) {
  int i = blockIdx.x * blockDim.x + threadIdx.x;
  if (i < n) S[i] /= fmaxf(deg[i / HID_C], 1.0f);
}

// per (node, head): al = <xh[n,h,:], a_src[h,:]>, ar = <xh[n,h,:], a_dst[h,:]>
__global__ void k_alar(const float* __restrict__ xh, const float* __restrict__ asrc,
                       const float* __restrict__ adst, float* __restrict__ al,
                       float* __restrict__ ar, int Nn) {
  int t = blockIdx.x * blockDim.x + threadIdx.x;
  if (t >= Nn * HEADS_C) return;
  int n = t / HEADS_C, h = t % HEADS_C;
  const float* x = xh + n * HID_C + h * DH_C;
  const float* as = asrc + h * DH_C;
  const float* ad = adst + h * DH_C;
  float sa = 0.f, sb = 0.f;
  #pragma unroll 4
  for (int d = 0; d < DH_C; ++d) { sa += x[d] * as[d]; sb += x[d] * ad[d]; }
  al[t] = sa; ar[t] = sb;
}

// init segment-max with the self-loop logit (every node has one)
__global__ void k_minit(const float* __restrict__ al, const float* __restrict__ ar,
                        unsigned* __restrict__ mbuf, int n4) {
  int t = blockIdx.x * blockDim.x + threadIdx.x;
  if (t < n4) mbuf[t] = fenc(leakyf(al[t] + ar[t]));
}

__global__ void k_mmax(const int* __restrict__ src, const int* __restrict__ dst,
                       const float* __restrict__ al, const float* __restrict__ ar,
                       unsigned* __restrict__ mbuf, int E) {
  int t = blockIdx.x * blockDim.x + threadIdx.x;
  if (t >= E * HEADS_C) return;
  int e = t >> 2, h = t & 3;
  int s = src[e], d = dst[e];
  float v = leakyf(al[s * HEADS_C + h] + ar[d * HEADS_C + h]);
  atomicMax(&mbuf[d * HEADS_C + h], fenc(v));
}

// decode max in place, seed denominator with self-loop exp
__global__ void k_mdec_den(const float* __restrict__ al, const float* __restrict__ ar,
                           unsigned* __restrict__ mbuf, float* __restrict__ den, int n4) {
  int t = blockIdx.x * blockDim.x + threadIdx.x;
  if (t >= n4) return;
  float m = fdec(mbuf[t]);
  float ex = __expf(leakyf(al[t] + ar[t]) - m);
  den[t] = ex;
  ((float*)mbuf)[t] = m;  // reuse as float max buffer
}

// num[n, :] = ex_self(n, head) * xh[n, :]
__global__ void k_numinit(const float* __restrict__ xh, const float* __restrict__ den,
                          float* __restrict__ num, int n) {
  int i = blockIdx.x * blockDim.x + threadIdx.x;
  if (i >= n) return;
  int nn = i / HID_C, h = (i % HID_C) / DH_C;
  num[i] = den[nn * HEADS_C + h] * xh[i];
}

// per (edge, head): ex = exp(e - m[dst]); stash; den[dst] += ex
__global__ void k_edge_ex(const int* __restrict__ src, const int* __restrict__ dst,
                          const float* __restrict__ al, const float* __restrict__ ar,
                          const float* __restrict__ mf, float* __restrict__ exb,
                          float* __restrict__ den, int E) {
  int t = blockIdx.x * blockDim.x + threadIdx.x;
  if (t >= E * HEADS_C) return;
  int e = t >> 2, h = t & 3;
  int s = src[e], d = dst[e];
  float v = leakyf(al[s * HEADS_C + h] + ar[d * HEADS_C + h]);
  float ex = __expf(v - mf[d * HEADS_C + h]);
  exb[t] = ex;
  atomicAdd(&den[d * HEADS_C + h], ex);
}

// per (edge, 4-channel chunk): num[dst, c] += ex * xh[src, c]
__global__ void k_edge_num(const int* __restrict__ src, const int* __restrict__ dst,
                           const float* __restrict__ exb, const float* __restrict__ xh,
                           float* __restrict__ num, int E) {
  int t = blockIdx.x * blockDim.x + threadIdx.x;
  if (t >= E * (HID_C / 4)) return;
  int e = t / (HID_C / 4), c4 = t % (HID_C / 4);
  int h = c4 / (DH_C / 4);
  float ex = exb[e * HEADS_C + h];
  const float4 v = *(const float4*)(xh + src[e] * HID_C + c4 * 4);
  float* o = num + dst[e] * HID_C + c4 * 4;
  atomicAdd(o + 0, ex * v.x);
  atomicAdd(o + 1, ex * v.y);
  atomicAdd(o + 2, ex * v.z);
  atomicAdd(o + 3, ex * v.w);
}

// out = relu(bn(num/den + b)); optionally mirror to embeddings output
__global__ void k_gat_epi(const float* __restrict__ num, const float* __restrict__ den,
                          const float* __restrict__ b, const float* __restrict__ gamma,
                          const float* __restrict__ beta, float* __restrict__ out,
                          float* __restrict__ emb, int n) {
  int i = blockIdx.x * blockDim.x + threadIdx.x;
  if (i >= n) return;
  int nn = i / HID_C, c = i % HID_C, h = c / DH_C;
  float o = num[i] / fmaxf(den[nn * HEADS_C + h], 1e-16f) + b[c];
  o = o * (gamma[c] * rsqrtf(1.0f + EPS_C)) + beta[c];
  o = fmaxf(o, 0.f);
  out[i] = o;
  if (emb) emb[i] = o;
}

// final 160 -> 2 classifier
__global__ void k_cls2(const float* __restrict__ hid, const float* __restrict__ w2,
                       const float* __restrict__ b2, float* __restrict__ out, int Nn) {
  int nidx = blockIdx.x * blockDim.x + threadIdx.x;
  if (nidx >= Nn) return;
  const float* hrow = hid + nidx * (HID_C / 2);
  float a0 = b2[0], a1 = b2[1];
  #pragma unroll 4
  for (int k = 0; k < HID_C / 2; ++k) {
    float hv = hrow[k];
    a0 += hv * w2[k * 2 + 0];
    a1 += hv * w2[k * 2 + 1];
  }
  out[nidx * 2 + 0] = a0;
  out[nidx * 2 + 1] = a1;
}

inline int cdiv(long long a, int b) { return (int)((a + b - 1) / b); }

}  // namespace

extern "C" void kernel_launch(void* const* d_in, const int* in_sizes, int n_in,
                              void* d_out, int out_size, void* d_ws, size_t ws_size,
                              hipStream_t stream) {
  const float* x       = (const float*)d_in[0];
  const int*   ei      = (const int*)  d_in[1];
  const float* w_in    = (const float*)d_in[2];
  const float* b_in    = (const float*)d_in[3];
  const float* s0_wl   = (const float*)d_in[4];
  const float* s0_bl   = (const float*)d_in[5];
  const float* s0_wr   = (const float*)d_in[6];
  const float* g1_w    = (const float*)d_in[7];
  const float* g1_as   = (const float*)d_in[8];
  const float* g1_ad   = (const float*)d_in[9];
  const float* g1_b    = (const float*)d_in[10];
  const float* s2_wl   = (const float*)d_in[11];
  const float* s2_bl   = (const float*)d_in[12];
  const float* s2_wr   = (const float*)d_in[13];
  const float* g3_w    = (const float*)d_in[14];
  const float* g3_as   = (const float*)d_in[15];
  const float* g3_ad   = (const float*)d_in[16];
  const float* g3_b    = (const float*)d_in[17];
  const float* bn_g    = (const float*)d_in[18];
  const float* bn_b    = (const float*)d_in[19];
  const float* c_w1    = (const float*)d_in[20];
  const float* c_b1    = (const float*)d_in[21];
  const float* c_w2    = (const float*)d_in[22];
  const float* c_b2    = (const float*)d_in[23];

  const int Nn = in_sizes[0] / IN_F;        // 50000
  const int E  = in_sizes[1] / 2;           // 800000
  const int* src = ei;
  const int* dst = ei + E;

  float* outp = (float*)d_out;              // [N,2] then [N,HID]
  float* emb_out = outp + (size_t)Nn * 2;

  // workspace layout
  float* H0  = (float*)d_ws;
  float* H1  = H0 + (size_t)Nn * HID_C;
  float* H2  = H1 + (size_t)Nn * HID_C;
  float* deg = H2 + (size_t)Nn * HID_C;
  float* al  = deg + Nn;
  float* ar  = al + (size_t)Nn * HEADS_C;
  float* mf  = ar + (size_t)Nn * HEADS_C;   // uint-encoded max, then decoded float
  float* den = mf + (size_t)Nn * HEADS_C;
  float* exb = den + (size_t)Nn * HEADS_C;  // E * HEADS

  const int TPB = 256;
  const int MT = Nn / 16;                   // 3125 M-tiles
  constexpr int NT = 5;                     // 16x80 strip per wave
  const dim3 b32(32);
  const dim3 gHID(MT, HID_C / (16 * NT));   // x 4
  const dim3 gCLS(MT, (HID_C / 2) / (16 * NT)); // x 2
  const int nNH  = Nn * HID_C;
  const int nN4  = Nn * HEADS_C;
  const int nE4  = E * HEADS_C;
  const int nE80 = E * (HID_C / 4);

  // degree (shared by both SAGE layers)
  k_fill<<<cdiv(Nn, TPB), TPB, 0, stream>>>(deg, 0.f, Nn);
  k_deg<<<cdiv(E, TPB), TPB, 0, stream>>>(dst, deg, E);

  // input MLP: H0 = relu(x @ w_in + b_in)
  k_gemm_bias_relu<NT><<<gHID, b32, 0, stream>>>(x, w_in, b_in, H0, IN_F, HID_C);

  // ---- layer 0: SAGE (cur=H0 -> H2) ----
  k_fill<<<cdiv(nNH, TPB), TPB, 0, stream>>>(H1, 0.f, nNH);
  k_scatter<<<cdiv(nE80, TPB), TPB, 0, stream>>>(H0, src, dst, H1, E);
  k_mean<<<cdiv(nNH, TPB), TPB, 0, stream>>>(H1, deg, nNH);
  k_gemm_sage<NT><<<gHID, b32, 0, stream>>>(H1, s0_wl, H0, s0_wr, s0_bl,
                                            bn_g + 0 * HID_C, bn_b + 0 * HID_C, H2);

  // ---- layer 1: GAT (cur=H2, xh->H0, num->H1, out->H2) ----
  k_gemm_plain<NT><<<gHID, b32, 0, stream>>>(H2, g1_w, H0, HID_C, HID_C);
  k_alar<<<cdiv(nN4, TPB), TPB, 0, stream>>>(H0, g1_as, g1_ad, al, ar, Nn);
  k_minit<<<cdiv(nN4, TPB), TPB, 0, stream>>>(al, ar, (unsigned*)mf, nN4);
  k_mmax<<<cdiv(nE4, TPB), TPB, 0, stream>>>(src, dst, al, ar, (unsigned*)mf, E);
  k_mdec_den<<<cdiv(nN4, TPB), TPB, 0, stream>>>(al, ar, (unsigned*)mf, den, nN4);
  k_numinit<<<cdiv(nNH, TPB), TPB, 0, stream>>>(H0, den, H1, nNH);
  k_edge_ex<<<cdiv(nE4, TPB), TPB, 0, stream>>>(src, dst, al, ar, mf, exb, den, E);
  k_edge_num<<<cdiv(nE80, TPB), TPB, 0, stream>>>(src, dst, exb, H0, H1, E);
  k_gat_epi<<<cdiv(nNH, TPB), TPB, 0, stream>>>(H1, den, g1_b, bn_g + 1 * HID_C,
                                                bn_b + 1 * HID_C, H2, nullptr, nNH);

  // ---- layer 2: SAGE (cur=H2 -> H0) ----
  k_fill<<<cdiv(nNH, TPB), TPB, 0, stream>>>(H1, 0.f, nNH);
  k_scatter<<<cdiv(nE80, TPB), TPB, 0, stream>>>(H2, src, dst, H1, E);
  k_mean<<<cdiv(nNH, TPB), TPB, 0, stream>>>(H1, deg, nNH);
  k_gemm_sage<NT><<<gHID, b32, 0, stream>>>(H1, s2_wl, H2, s2_wr, s2_bl,
                                            bn_g + 2 * HID_C, bn_b + 2 * HID_C, H0);

  // ---- layer 3: GAT (cur=H0, xh->H1, num->H2, out->H0 + embeddings) ----
  k_gemm_plain<NT><<<gHID, b32, 0, stream>>>(H0, g3_w, H1, HID_C, HID_C);
  k_alar<<<cdiv(nN4, TPB), TPB, 0, stream>>>(H1, g3_as, g3_ad, al, ar, Nn);
  k_minit<<<cdiv(nN4, TPB), TPB, 0, stream>>>(al, ar, (unsigned*)mf, nN4);
  k_mmax<<<cdiv(nE4, TPB), TPB, 0, stream>>>(src, dst, al, ar, (unsigned*)mf, E);
  k_mdec_den<<<cdiv(nN4, TPB), TPB, 0, stream>>>(al, ar, (unsigned*)mf, den, nN4);
  k_numinit<<<cdiv(nNH, TPB), TPB, 0, stream>>>(H1, den, H2, nNH);
  k_edge_ex<<<cdiv(nE4, TPB), TPB, 0, stream>>>(src, dst, al, ar, mf, exb, den, E);
  k_edge_num<<<cdiv(nE80, TPB), TPB, 0, stream>>>(src, dst, exb, H1, H2, E);
  k_gat_epi<<<cdiv(nNH, TPB), TPB, 0, stream>>>(H2, den, g3_b, bn_g + 3 * HID_C,
                                                bn_b + 3 * HID_C, H0, emb_out, nNH);

  // ---- classifier ----
  k_gemm_bias_relu<NT><<<gCLS, b32, 0, stream>>>(H0, c_w1, c_b1, H1, HID_C, HID_C / 2);
  k_cls2<<<cdiv(Nn, TPB), TPB, 0, stream>>>(H1, c_w2, c_b2, outp, Nn);
}